// OntologyNN_26139170963549
// MI455X (gfx1250) — compile-verified
//
#include <hip/hip_runtime.h>
#include <math.h>

// fp32 WMMA fragments on gfx1250 (wave32):
//  A 16x4 f32  -> 2 VGPRs/lane  : lane m=lane&15 holds A[m][k0], A[m][k0+1], k0 = kb + 2*(lane>>4)
//  B 4x16 f32  -> 2 VGPRs/lane  : lane n=lane&15 holds B[k0][n], B[k0+1][n]
//  C/D 16x16   -> 8 VGPRs/lane  : vgpr r, lane -> D[r + 8*(lane>>4)][lane&15]
typedef __attribute__((ext_vector_type(2))) float v2f;
typedef __attribute__((ext_vector_type(8))) float v8f;

#define EPS_BN 1e-5f
#define XDIM   1168
#define NWAVES 2      // waves per block; each wave owns a 16-row batch tile

__device__ __forceinline__ void lds_fence() {
  // Intra-wave producer->consumer through LDS: stop compiler reordering and
  // drain DScnt (LDS is in-order within a wave; no cross-wave sharing here).
  asm volatile("s_wait_dscnt 0" ::: "memory");
}

__device__ __forceinline__ v8f wmma_f32_k4(v2f a, v2f b, v8f c) {
  // D = A(16x4,f32) x B(4x16,f32) + C(16x16,f32)
  return __builtin_amdgcn_wmma_f32_16x16x4_f32(false, a, false, b, (short)0, c,
                                               false, false);
}

// CDNA5 has hardware v_tanh_f32; use it if the toolchain exposes the builtin,
// otherwise a branch-free 3-op approximation via v_exp_f32 + v_rcp_f32:
//   tanh(x) = 1 - 2/(exp2(2*log2(e)*x) + 1)   (saturates correctly at +-inf)
__device__ __forceinline__ float fast_tanh(float x) {
#if __has_builtin(__builtin_amdgcn_tanhf)
  return __builtin_amdgcn_tanhf(x);
#elif __has_builtin(__builtin_amdgcn_tanh_f32)
  return __builtin_amdgcn_tanh_f32(x);
#else
  float e = __builtin_amdgcn_exp2f(2.8853900817779268f * x);
  return 1.0f - 2.0f * __builtin_amdgcn_rcpf(e + 1.0f);
#endif
}

__global__ __launch_bounds__(NWAVES * 32)
void ontology_fused_kernel(
    const float* __restrict__ x,
    const float* __restrict__ Wl, const float* __restrict__ bl,
    const float* __restrict__ gl, const float* __restrict__ betal,
    const float* __restrict__ ml, const float* __restrict__ vl,
    const float* __restrict__ Wm, const float* __restrict__ bm,
    const float* __restrict__ gm, const float* __restrict__ betam,
    const float* __restrict__ mm, const float* __restrict__ vm,
    const float* __restrict__ Wr, const float* __restrict__ br,
    const float* __restrict__ gr, const float* __restrict__ betar,
    const float* __restrict__ mr, const float* __restrict__ vr,
    const float* __restrict__ Wh, const float* __restrict__ bh,
    float* __restrict__ out, int n_rows)
{
  __shared__ float s_mid [NWAVES][16 * 144];  // mid input staging (reused as 16x64 root hidden)
  __shared__ float s_root[NWAVES][16 * 272];  // root input staging

  const int wave = threadIdx.x >> 5;
  const int lane = threadIdx.x & 31;
  const int mn   = lane & 15;   // A row index == B/C/D column index for this lane
  const int kh   = lane >> 4;   // K-half / M-half selector

  float* midbuf  = s_mid [wave];
  float* rootbuf = s_root[wave];

  const long row0 = ((long)blockIdx.x * NWAVES + wave) * 16;
  if (row0 >= n_rows) return;
  const float* xt = x + row0 * XDIM;

  // ---------------- per-mid group: 8 leaves, then the mid node ----------------
  for (int g = 0; g < 8; ++g) {
    // stage mid own features into mid_in[:, 0:16]  (x is streamed once: NT loads)
    for (int i = 0; i < 8; ++i) {
      int idx = lane + 32 * i;               // 0..255
      int r = idx >> 4, c = idx & 15;
      midbuf[r * 144 + c] =
          __builtin_nontemporal_load(xt + (long)r * XDIM + 1024 + g * 16 + c);
    }

    // 8 leaf nodes: [16x16] @ [16x16], BN folded, tanh, write to mid_in[:, 16+j*16 ...]
    for (int j = 0; j < 8; ++j) {
      const int l  = g * 8 + j;
      const int li = l * 16 + mn;
      const float inv = rsqrtf(vl[li] + EPS_BN);
      const float sc  = gl[li] * inv;
      const float of  = (bl[li] - ml[li]) * sc + betal[li];
      v8f acc;
      #pragma unroll
      for (int r = 0; r < 8; ++r) acc[r] = of;
      const float* Wn = Wl + (l * 16) * 16 + mn;   // W[k][n] at Wn[k*16]
      #pragma unroll
      for (int kb = 0; kb < 16; kb += 4) {
        const int k0 = kb + 2 * kh;
        v2f a = __builtin_nontemporal_load(
            (const v2f*)(xt + (long)mn * XDIM + l * 16 + k0));
        v2f b;
        b.x = Wn[ k0      * 16] * sc;
        b.y = Wn[(k0 + 1) * 16] * sc;
        acc = wmma_f32_k4(a, b, acc);
      }
      #pragma unroll
      for (int r = 0; r < 8; ++r)
        midbuf[(r + 8 * kh) * 144 + 16 + j * 16 + mn] = fast_tanh(acc[r]);
    }
    lds_fence();

    // mid node: [16x144] @ [144x32] -> two 16x16 N-tiles, write into root_in
    for (int nt = 0; nt < 2; ++nt) {
      const int ng = nt * 16 + mn;
      const int gi = g * 32 + ng;
      const float inv = rsqrtf(vm[gi] + EPS_BN);
      const float sc  = gm[gi] * inv;
      const float of  = (bm[gi] - mm[gi]) * sc + betam[gi];
      v8f acc;
      #pragma unroll
      for (int r = 0; r < 8; ++r) acc[r] = of;
      const float* Wn = Wm + (g * 144) * 32 + ng;
      for (int kb = 0; kb < 144; kb += 4) {
        const int k0 = kb + 2 * kh;
        v2f a = *(const v2f*)(midbuf + mn * 144 + k0);
        v2f b;
        b.x = Wn[ k0      * 32] * sc;
        b.y = Wn[(k0 + 1) * 32] * sc;
        acc = wmma_f32_k4(a, b, acc);
      }
      #pragma unroll
      for (int r = 0; r < 8; ++r)
        rootbuf[(r + 8 * kh) * 272 + 16 + g * 32 + ng] = fast_tanh(acc[r]);
    }
    lds_fence();   // mid-stage LDS reads done before next group's staging stores
  }

  // stage root own features into root_in[:, 0:16]
  for (int i = 0; i < 8; ++i) {
    int idx = lane + 32 * i;
    int r = idx >> 4, c = idx & 15;
    rootbuf[r * 272 + c] =
        __builtin_nontemporal_load(xt + (long)r * XDIM + 1152 + c);
  }
  lds_fence();

  // root: [16x272] @ [272x64] -> 4 N-tiles; root hidden reuses midbuf as 16x64
  for (int nt = 0; nt < 4; ++nt) {
    const int ng = nt * 16 + mn;
    const float inv = rsqrtf(vr[ng] + EPS_BN);
    const float sc  = gr[ng] * inv;
    const float of  = (br[ng] - mr[ng]) * sc + betar[ng];
    v8f acc;
    #pragma unroll
    for (int r = 0; r < 8; ++r) acc[r] = of;
    const float* Wn = Wr + ng;                     // Wr[k*64 + ng]
    for (int kb = 0; kb < 272; kb += 4) {
      const int k0 = kb + 2 * kh;
      v2f a = *(const v2f*)(rootbuf + mn * 272 + k0);
      v2f b;
      b.x = Wn[ k0      * 64] * sc;
      b.y = Wn[(k0 + 1) * 64] * sc;
      acc = wmma_f32_k4(a, b, acc);
    }
    #pragma unroll
    for (int r = 0; r < 8; ++r)
      midbuf[(r + 8 * kh) * 64 + nt * 16 + mn] = fast_tanh(acc[r]);
  }
  lds_fence();

  // head: [16x64] @ [64x10] + bh, N padded to 16 with zero columns
  {
    const float of = (mn < 10) ? bh[mn] : 0.0f;
    v8f acc;
    #pragma unroll
    for (int r = 0; r < 8; ++r) acc[r] = of;
    for (int kb = 0; kb < 64; kb += 4) {
      const int k0 = kb + 2 * kh;
      v2f a = *(const v2f*)(midbuf + mn * 64 + k0);
      v2f b;
      b.x = (mn < 10) ? Wh[ k0      * 10 + mn] : 0.0f;
      b.y = (mn < 10) ? Wh[(k0 + 1) * 10 + mn] : 0.0f;
      acc = wmma_f32_k4(a, b, acc);   // EXEC all-ones; masking only on loads/stores
    }
    if (mn < 10) {
      #pragma unroll
      for (int r = 0; r < 8; ++r)
        __builtin_nontemporal_store(acc[r],
                                    out + (row0 + r + 8 * kh) * 10 + mn);
    }
  }
}

extern "C" void kernel_launch(void* const* d_in, const int* in_sizes, int n_in,
                              void* d_out, int out_size, void* d_ws, size_t ws_size,
                              hipStream_t stream) {
  const float* x     = (const float*)d_in[0];
  const float* Wl    = (const float*)d_in[1];
  const float* bl    = (const float*)d_in[2];
  const float* gl    = (const float*)d_in[3];
  const float* betal = (const float*)d_in[4];
  const float* ml    = (const float*)d_in[5];
  const float* vl    = (const float*)d_in[6];
  const float* Wm    = (const float*)d_in[7];
  const float* bm    = (const float*)d_in[8];
  const float* gm    = (const float*)d_in[9];
  const float* betam = (const float*)d_in[10];
  const float* mm    = (const float*)d_in[11];
  const float* vm    = (const float*)d_in[12];
  const float* Wr    = (const float*)d_in[13];
  const float* br    = (const float*)d_in[14];
  const float* gr    = (const float*)d_in[15];
  const float* betar = (const float*)d_in[16];
  const float* mr    = (const float*)d_in[17];
  const float* vr    = (const float*)d_in[18];
  const float* Wh    = (const float*)d_in[19];
  const float* bh    = (const float*)d_in[20];

  const int n_rows = in_sizes[0] / XDIM;                 // 16384
  const int blocks = (n_rows + NWAVES * 16 - 1) / (NWAVES * 16);

  ontology_fused_kernel<<<blocks, NWAVES * 32, 0, stream>>>(
      x, Wl, bl, gl, betal, ml, vl, Wm, bm, gm, betam, mm, vm,
      Wr, br, gr, betar, mr, vr, Wh, bh, (float*)d_out, n_rows);
}